// RaggedGravNet_simple_58325655880002
// MI455X (gfx1250) — compile-verified
//
#include <hip/hip_runtime.h>

typedef float v2f __attribute__((ext_vector_type(2)));
typedef float v8f __attribute__((ext_vector_type(8)));

constexpr int N_PTS  = 32768;
constexpr int SEGSZ  = 4096;
constexpr int KNN    = 39;     // K=40 incl. self, self dropped
constexpr int F_INC  = 64;     // F_IN
constexpr int NPROP  = 64;     // n_propagate
constexpr int NF     = 128;    // filters per dense layer

static __device__ __forceinline__ v8f wmma_f32_16x16x4(v2f a, v2f b, v8f c) {
  // fp32 WMMA: D = A(16x4) * B(4x16) + C(16x16)
  return __builtin_amdgcn_wmma_f32_16x16x4_f32(false, a, false, b, (short)0, c,
                                               false, false);
}

// ---------------------------------------------------------------------------
// Weight repack: Wt[c*fi + k] = W[k*cols + c]  (transpose so each lane's
// (k,k+1) B pair is one aligned b64 load in the WMMA loops)
// ---------------------------------------------------------------------------
__global__ void k_repack(const float* __restrict__ W, float* __restrict__ Wt,
                         int fi, int cols) {
  const int idx = blockIdx.x * 256 + threadIdx.x;
  if (idx < fi * cols) {
    const int c = idx / fi;
    const int k = idx % fi;
    Wt[idx] = W[k * cols + c];
  }
}

// Ws (64x4) -> padded transposed 16x64 (cols 4..15 zero) so the coords GEMM
// B operand needs no exec-divergent load.
__global__ void k_repack_ws(const float* __restrict__ Ws, float* __restrict__ Wsp) {
  const int idx = blockIdx.x * 256 + threadIdx.x;
  if (idx < 16 * 64) {
    const int c = idx / 64;
    const int k = idx % 64;
    Wsp[idx] = (c < 4) ? Ws[k * 4 + c] : 0.0f;
  }
}

// ---------------------------------------------------------------------------
// Kernel 1: coords = x@Ws + bs  (N x 4),  feat0 = x@Wf + bf  (N x 64)
// One wave handles a 16-row tile; K loop over 64 in steps of 4.
// ---------------------------------------------------------------------------
__global__ void __launch_bounds__(256)
k_input(const float* __restrict__ x, const float* __restrict__ Wsp,
        const float* __restrict__ bsv, const float* __restrict__ Wft,
        const float* __restrict__ bfv, float* __restrict__ coords,
        float* __restrict__ feat) {
  const int lane = threadIdx.x & 31;
  const int wv   = threadIdx.x >> 5;
  const int half = lane >> 4;     // 0: K pair {0,1}, 1: K pair {2,3}
  const int nl   = lane & 15;     // A: row M; B/C: column N
  const int row0 = blockIdx.x * 128 + wv * 16;

  v8f cc = {};
  v8f cf[4];
#pragma unroll
  for (int t = 0; t < 4; ++t) cf[t] = cc;

  for (int k = 0; k < F_INC; k += 4) {
    const int kk = k + 2 * half;
    const v2f a = *(const v2f*)&x[(size_t)(row0 + nl) * F_INC + kk];
    // load all B operands first (one clause), then issue the 5 WMMAs
    const v2f bc = *(const v2f*)&Wsp[(size_t)nl * 64 + kk];
    v2f bw[4];
#pragma unroll
    for (int t = 0; t < 4; ++t)
      bw[t] = *(const v2f*)&Wft[(size_t)(t * 16 + nl) * 64 + kk];
    cc = wmma_f32_16x16x4(a, bc, cc);
#pragma unroll
    for (int t = 0; t < 4; ++t) cf[t] = wmma_f32_16x16x4(a, bw[t], cf[t]);
  }
#pragma unroll
  for (int r = 0; r < 8; ++r) {
    const int row = row0 + r + 8 * half;
    if (nl < 4) coords[(size_t)row * 4 + nl] = cc[r] + bsv[nl];
#pragma unroll
    for (int t = 0; t < 4; ++t)
      feat[(size_t)row * NPROP + t * 16 + nl] = cf[t][r] + bfv[t * 16 + nl];
  }
}

// ---------------------------------------------------------------------------
// Kernel 2: brute-force kNN per segment.
// d2(q,c) = |q|^2 + |c|^2 - 2 q.c ; inner products via rank-4 fp32 WMMA.
// Candidate tiles double-buffered in LDS via async global->LDS loads, so the
// ASYNCcnt-tracked copy of tile t+1 overlaps the WMMA + top-k scan of tile t.
// ---------------------------------------------------------------------------
__global__ void __launch_bounds__(256)
k_knn(const float* __restrict__ coords, int* __restrict__ nidx,
      float* __restrict__ wgt) {
  const int lane = threadIdx.x & 31;
  const int wv   = threadIdx.x >> 5;
  const int half = lane >> 4;
  const int nl   = lane & 15;
  const int q0   = blockIdx.x * 128 + wv * 16;
  const int segBase = (blockIdx.x / (SEGSZ / 128)) * SEGSZ;

  __shared__ float s_cand[2][64 * 4];
  __shared__ float s_cn2[64];
  __shared__ float s_qn2[8][16];
  __shared__ float s_d2[8][16][64];

  // flat->LDS offset: hardware takes addr[31:0] for the LDS aperture
  const unsigned lds_off0 = (unsigned)(uintptr_t)(void*)&s_cand[0][threadIdx.x];
  const unsigned lds_off1 = (unsigned)(uintptr_t)(void*)&s_cand[1][threadIdx.x];

  auto issue_tile = [&](int tile, int b) {
    const unsigned long long ga = (unsigned long long)(uintptr_t)
        &coords[((size_t)segBase + (size_t)tile * 64) * 4 + threadIdx.x];
    const unsigned lo = b ? lds_off1 : lds_off0;
    asm volatile("global_load_async_to_lds_b32 %0, %1, off"
                 :: "v"(lo), "v"(ga) : "memory");
  };

  // A operand: this wave's 16 query coords (held for the whole segment scan)
  v2f a;
  a.x = coords[(size_t)(q0 + nl) * 4 + 2 * half];
  a.y = coords[(size_t)(q0 + nl) * 4 + 2 * half + 1];
  if (lane < 16) {
    const float* qc = &coords[(size_t)(q0 + lane) * 4];
    s_qn2[wv][lane] = qc[0]*qc[0] + qc[1]*qc[1] + qc[2]*qc[2] + qc[3]*qc[3];
  }

  float bd[KNN];
  int   bi[KNN];
  for (int i = 0; i < KNN; ++i) { bd[i] = 3.4e38f; bi[i] = segBase; }

  issue_tile(0, 0);  // prefetch first candidate tile
  __syncthreads();   // s_qn2 visible

  int buf = 0;
  for (int tile = 0; tile < SEGSZ / 64; ++tile) {
    const int cb = segBase + tile * 64;
    asm volatile("s_wait_asynccnt 0x0" ::: "memory");  // tile data in LDS
    __syncthreads();  // every wave's async slice landed; prev readers done
    if (tile + 1 < SEGSZ / 64) issue_tile(tile + 1, buf ^ 1);  // overlap copy
    if (threadIdx.x < 64) {
      const float* cc2 = &s_cand[buf][threadIdx.x * 4];
      s_cn2[threadIdx.x] = cc2[0]*cc2[0] + cc2[1]*cc2[1] + cc2[2]*cc2[2] + cc2[3]*cc2[3];
    }
    __syncthreads();
#pragma unroll
    for (int n0 = 0; n0 < 64; n0 += 16) {
      v2f b;
      b.x = s_cand[buf][(n0 + nl) * 4 + 2 * half];
      b.y = s_cand[buf][(n0 + nl) * 4 + 2 * half + 1];
      v8f ip = {};
      ip = wmma_f32_16x16x4(a, b, ip);
#pragma unroll
      for (int r = 0; r < 8; ++r) {
        const int m = r + 8 * half;
        s_d2[wv][m][n0 + nl] = s_qn2[wv][m] + s_cn2[n0 + nl] - 2.0f * ip[r];
      }
    }
    __syncthreads();
    if (lane < 16) {
      const int q = q0 + lane;
      for (int n = 0; n < 64; ++n) {
        const int j = cb + n;
        const float d = s_d2[wv][lane][n];
        if (j != q && d < bd[KNN - 1]) {
          int p = KNN - 1;
          while (p > 0 && bd[p - 1] > d) {
            bd[p] = bd[p - 1]; bi[p] = bi[p - 1]; --p;
          }
          bd[p] = d; bi[p] = j;
        }
      }
    }
    buf ^= 1;
  }

  if (lane < 16) {
    const int q = q0 + lane;
    for (int k = 0; k < KNN; ++k) {
      nidx[(size_t)q * KNN + k] = bi[k];
      wgt[(size_t)q * KNN + k]  = expf(-(bd[k] * 10.0f + 1e-5f));
    }
  }
}

// ---------------------------------------------------------------------------
// Kernel 3: agg[n] = [ max_k feat[nb_k]*w_k , mean_k feat[nb_k]*w_k ]
// One wave per node; neighbor ids/weights staged in LDS; each lane covers two
// adjacent features so every gather is one aligned b64 load.
// ---------------------------------------------------------------------------
__global__ void __launch_bounds__(256)
k_aggregate(const float* __restrict__ feat, const int* __restrict__ nidx,
            const float* __restrict__ wgt, float* __restrict__ agg, int F) {
  const int lane = threadIdx.x & 31;
  const int wv   = threadIdx.x >> 5;
  const int node = blockIdx.x * 8 + wv;

  __shared__ int   s_nb[8][40];
  __shared__ float s_w[8][40];
  s_nb[wv][lane] = nidx[(size_t)node * KNN + lane];
  s_w[wv][lane]  = wgt[(size_t)node * KNN + lane];
  if (lane < KNN - 32) {
    s_nb[wv][32 + lane] = nidx[(size_t)node * KNN + 32 + lane];
    s_w[wv][32 + lane]  = wgt[(size_t)node * KNN + 32 + lane];
  }
  __syncthreads();

  const int twoF = 2 * F;
  for (int f0 = lane * 2; f0 < F; f0 += 64) {
    float mx0 = -3.4e38f, mx1 = -3.4e38f, sm0 = 0.0f, sm1 = 0.0f;
    for (int k = 0; k < KNN; ++k) {
      const v2f v  = *(const v2f*)&feat[(size_t)s_nb[wv][k] * F + f0];
      const float wk = s_w[wv][k];
      const float a0 = v.x * wk, a1 = v.y * wk;
      mx0 = fmaxf(mx0, a0); mx1 = fmaxf(mx1, a1);
      sm0 += a0;            sm1 += a1;
    }
    v2f omax = {mx0, mx1};
    v2f omean = {sm0 * (1.0f / KNN), sm1 * (1.0f / KNN)};
    *(v2f*)&agg[(size_t)node * twoF + f0]     = omax;
    *(v2f*)&agg[(size_t)node * twoF + F + f0] = omean;
  }
}

// ---------------------------------------------------------------------------
// Kernel 4: out = tanh(concat([x, agg]) @ W + b), Wt is transposed [128, fi].
// One wave computes a 16x128 output tile (8 fp32 WMMA accumulators).
// ---------------------------------------------------------------------------
__global__ void __launch_bounds__(256)
k_dense(const float* __restrict__ xin, const float* __restrict__ agg,
        int aggCols, const float* __restrict__ Wt,
        const float* __restrict__ bias, float* __restrict__ out, int fi) {
  const int lane = threadIdx.x & 31;
  const int wv   = threadIdx.x >> 5;
  const int half = lane >> 4;
  const int nl   = lane & 15;
  const int row0 = blockIdx.x * 128 + wv * 16;
  const int rowA = row0 + nl;

  v8f c[8];
#pragma unroll
  for (int t = 0; t < 8; ++t) c[t] = (v8f){};

  for (int k = 0; k < fi; k += 4) {
    const int kk = k + 2 * half;
    // concat([x, agg]) A fetch — region is uniform per K-chunk (64 % 4 == 0)
    const float* asrc = (kk < F_INC)
                            ? &xin[(size_t)rowA * F_INC + kk]
                            : &agg[(size_t)rowA * aggCols + (kk - F_INC)];
    const v2f a = *(const v2f*)asrc;
    v2f b[8];
#pragma unroll
    for (int t = 0; t < 8; ++t)
      b[t] = *(const v2f*)&Wt[(size_t)(t * 16 + nl) * fi + kk];
#pragma unroll
    for (int t = 0; t < 8; ++t) c[t] = wmma_f32_16x16x4(a, b[t], c[t]);
  }
#pragma unroll
  for (int r = 0; r < 8; ++r) {
    const int row = row0 + r + 8 * half;
#pragma unroll
    for (int t = 0; t < 8; ++t)
      out[(size_t)row * NF + t * 16 + nl] = tanhf(c[t][r] + bias[t * 16 + nl]);
  }
}

// ---------------------------------------------------------------------------
extern "C" void kernel_launch(void* const* d_in, const int* in_sizes, int n_in,
                              void* d_out, int out_size, void* d_ws, size_t ws_size,
                              hipStream_t stream) {
  const float* x  = (const float*)d_in[0];
  // d_in[1] = row_splits (int64), uniform segments — unused on device
  const float* Ws = (const float*)d_in[2];
  const float* bs = (const float*)d_in[3];
  const float* Wf = (const float*)d_in[4];
  const float* bf = (const float*)d_in[5];
  const float* W0 = (const float*)d_in[6];
  const float* b0 = (const float*)d_in[7];
  const float* W1 = (const float*)d_in[8];
  const float* b1 = (const float*)d_in[9];
  (void)in_sizes; (void)n_in; (void)out_size; (void)ws_size;

  char* ws = (char*)d_ws;
  size_t off = 0;
  float* coords = (float*)(ws + off); off += (size_t)N_PTS * 4 * sizeof(float);
  float* feat0  = (float*)(ws + off); off += (size_t)N_PTS * NPROP * sizeof(float);
  int*   nidx   = (int*)  (ws + off); off += (size_t)N_PTS * KNN * sizeof(int);
  float* wgt    = (float*)(ws + off); off += (size_t)N_PTS * KNN * sizeof(float);
  float* aggb   = (float*)(ws + off); off += (size_t)N_PTS * 256 * sizeof(float);
  float* feat1  = (float*)(ws + off); off += (size_t)N_PTS * NF * sizeof(float);
  float* Wsp    = (float*)(ws + off); off += (size_t)16 * 64 * sizeof(float);
  float* Wft    = (float*)(ws + off); off += (size_t)64 * 64 * sizeof(float);
  float* W0t    = (float*)(ws + off); off += (size_t)128 * 192 * sizeof(float);
  float* W1t    = (float*)(ws + off); off += (size_t)128 * 320 * sizeof(float);

  const dim3 blk(256);
  const int fi0 = F_INC + 2 * NPROP;   // 192
  const int fi1 = F_INC + 2 * NF;      // 320

  k_repack_ws<<<(16 * 64 + 255) / 256, blk, 0, stream>>>(Ws, Wsp);
  k_repack   <<<(64 * 64 + 255) / 256, blk, 0, stream>>>(Wf, Wft, 64, 64);
  k_repack   <<<(fi0 * NF + 255) / 256, blk, 0, stream>>>(W0, W0t, fi0, NF);
  k_repack   <<<(fi1 * NF + 255) / 256, blk, 0, stream>>>(W1, W1t, fi1, NF);

  k_input    <<<N_PTS / 128, blk, 0, stream>>>(x, Wsp, bs, Wft, bf, coords, feat0);
  k_knn      <<<N_PTS / 128, blk, 0, stream>>>(coords, nidx, wgt);
  k_aggregate<<<N_PTS / 8,   blk, 0, stream>>>(feat0, nidx, wgt, aggb, NPROP);
  k_dense    <<<N_PTS / 128, blk, 0, stream>>>(x, aggb, 2 * NPROP, W0t, b0, feat1, fi0);
  k_aggregate<<<N_PTS / 8,   blk, 0, stream>>>(feat1, nidx, wgt, aggb, NF);
  k_dense    <<<N_PTS / 128, blk, 0, stream>>>(x, aggb, 2 * NF, W1t, b1,
                                               (float*)d_out, fi1);
}